// SubjectLinear_42975442764242
// MI455X (gfx1250) — compile-verified
//
#include <hip/hip_runtime.h>

#define NUM_SUBJECTS 32
#define IN_DIM 128
#define OUT_DIM 128
#define BATCH 16384
#define TILE_STRIDE 8    // row-tile grid stride per subject

typedef __attribute__((ext_vector_type(2))) float v2f;
typedef __attribute__((ext_vector_type(8))) float v8f;

struct Ws {
    int counts[NUM_SUBJECTS];
    int cursor[NUM_SUBJECTS];
    int offsets[NUM_SUBJECTS + 1];
    int order[BATCH];
};

// ---- pass 1: zero counters (must run every launch; ws is not re-poisoned) ----
__global__ void k_zero(int* counts, int* cursor) {
    int t = threadIdx.x;
    if (t < NUM_SUBJECTS) { counts[t] = 0; cursor[t] = 0; }
}

// ---- pass 2: histogram over expert ids ----
__global__ void k_hist(const int* __restrict__ ixs, int* counts) {
    int s = blockIdx.x * blockDim.x + threadIdx.x;
    if (s < BATCH) atomicAdd(&counts[ixs[s]], 1);
}

// ---- pass 3: exclusive scan (32 entries, trivial) ----
__global__ void k_scan(const int* __restrict__ counts, int* offsets) {
    if (threadIdx.x == 0) {
        int acc = 0;
        for (int i = 0; i < NUM_SUBJECTS; ++i) { offsets[i] = acc; acc += counts[i]; }
        offsets[NUM_SUBJECTS] = acc;
    }
}

// ---- pass 4: bucket sample ids per expert ----
__global__ void k_scatter(const int* __restrict__ ixs, const int* __restrict__ offsets,
                          int* cursor, int* order) {
    int s = blockIdx.x * blockDim.x + threadIdx.x;
    if (s < BATCH) {
        int g = ixs[s];
        int p = atomicAdd(&cursor[g], 1);
        order[offsets[g] + p] = s;
    }
}

// ---- pass 5: grouped GEMM via V_WMMA_F32_16X16X4_F32 ----
// block = 256 threads = 8 waves; wave w handles output columns [16w, 16w+16).
// B tile (16 cols x 128 K) is held in 32 v2f register fragments, loaded once
// per block and reused across all row tiles this block processes.
__global__ __launch_bounds__(256) void k_gemm(
    const float* __restrict__ X,      // [BATCH, IN_DIM]
    const float* __restrict__ W,      // [NUM_SUBJECTS, IN_DIM, OUT_DIM]
    const float* __restrict__ Bv,     // [NUM_SUBJECTS, OUT_DIM]
    const int*   __restrict__ counts,
    const int*   __restrict__ offsets,
    const int*   __restrict__ order,
    float*       __restrict__ out)    // [BATCH, OUT_DIM]
{
    const int g    = blockIdx.x & (NUM_SUBJECTS - 1);
    const int t0   = blockIdx.x >> 5;
    const int lane = threadIdx.x & 31;
    const int wid  = threadIdx.x >> 5;
    const int n0   = wid * 16;

    const int cnt = counts[g];
    if (cnt == 0) return;
    const int off       = offsets[g];
    const int fullTiles = cnt >> 4;
    const int ntiles    = (cnt + 15) >> 4;
    const int lastIdx   = off + cnt - 1;

    const int nlo   = lane & 15;          // 0..15
    const int hi    = lane >> 4;          // 0 for lanes 0-15, 1 for lanes 16-31
    const int khalf = hi * 2;             // K offset within a 4-wide step
    const int n     = n0 + nlo;           // output column handled by this lane

    const float* Wg = W + (size_t)g * IN_DIM * OUT_DIM;
    const float bv  = Bv[g * OUT_DIM + n];

    // ---- hoist B: 32 fragments, K-steps of 4. VGPR0=W[k][n], VGPR1=W[k+1][n]
    v2f bfrag[IN_DIM / 4];
    #pragma unroll
    for (int kk = 0; kk < IN_DIM / 4; ++kk) {
        const int k = kk * 4 + khalf;
        bfrag[kk].x = Wg[(k    ) * OUT_DIM + n];
        bfrag[kk].y = Wg[(k + 1) * OUT_DIM + n];
    }

    for (int t = t0; t < ntiles; t += TILE_STRIDE) {
        const int base = off + t * 16;

        // A gather: clamp index into the bucket; padded rows only pollute
        // output rows we never store, so no masking is needed.
        int idxA = base + nlo;
        idxA = idxA > lastIdx ? lastIdx : idxA;
        const int rowA  = order[idxA];
        const float* xp = X + (size_t)rowA * IN_DIM + khalf;

        v8f c = {};
        #pragma unroll
        for (int kk = 0; kk < IN_DIM / 4; ++kk) {
            // A 16x4 fragment: contiguous b64 per lane
            v2f a;
            a.x = xp[kk * 4];
            a.y = xp[kk * 4 + 1];
            c = __builtin_amdgcn_wmma_f32_16x16x4_f32(
                    false, a, false, bfrag[kk], (short)0, c, false, false);
        }

        // C/D layout: VGPR i -> row i (lanes 0-15) / row i+8 (lanes 16-31)
        if (t < fullTiles) {
            #pragma unroll
            for (int i = 0; i < 8; ++i) {
                const int m   = i + hi * 8;
                const int row = order[base + m];
                out[(size_t)row * OUT_DIM + n] = c[i] + bv;
            }
        } else {
            #pragma unroll
            for (int i = 0; i < 8; ++i) {
                const int m = i + hi * 8;
                if (t * 16 + m < cnt) {
                    const int row = order[base + m];
                    out[(size_t)row * OUT_DIM + n] = c[i] + bv;
                }
            }
        }
    }
}

extern "C" void kernel_launch(void* const* d_in, const int* in_sizes, int n_in,
                              void* d_out, int out_size, void* d_ws, size_t ws_size,
                              hipStream_t stream) {
    const float* x   = (const float*)d_in[0];
    const int*   ixs = (const int*)d_in[1];
    const float* w   = (const float*)d_in[2];
    const float* b   = (const float*)d_in[3];
    float* out = (float*)d_out;
    Ws* ws = (Ws*)d_ws;

    k_zero<<<1, 32, 0, stream>>>(ws->counts, ws->cursor);
    k_hist<<<BATCH / 256, 256, 0, stream>>>(ixs, ws->counts);
    k_scan<<<1, 32, 0, stream>>>(ws->counts, ws->offsets);
    k_scatter<<<BATCH / 256, 256, 0, stream>>>(ixs, ws->offsets, ws->cursor, ws->order);
    k_gemm<<<NUM_SUBJECTS * TILE_STRIDE, 256, 0, stream>>>(
        x, w, b, ws->counts, ws->offsets, ws->order, out);
}